// GCN_47742856462981
// MI455X (gfx1250) — compile-verified
//
#include <hip/hip_runtime.h>

#define N_NODES 100000
#define N_EDGES 1600000
#define DCH 128
#define NLAYERS 3
#define BN_EPS 1e-5f

#define SCAN_BLOCKS ((N_NODES + 255) / 256)   // 391

typedef float v2f __attribute__((ext_vector_type(2)));
typedef float v8f __attribute__((ext_vector_type(8)));

// ---------------- degree / in-degree count ----------------

__global__ void init_kernel(float* __restrict__ deg, int* __restrict__ cnt) {
    int i = blockIdx.x * blockDim.x + threadIdx.x;
    if (i < N_NODES) { deg[i] = 1.0f; cnt[i] = 0; }   // deg starts at self-loop weight
}

__global__ void deg_count_kernel(const int* __restrict__ dst,
                                 const float* __restrict__ ew,
                                 float* __restrict__ deg,
                                 int* __restrict__ cnt) {
    int e = blockIdx.x * blockDim.x + threadIdx.x;
    if (e < N_EDGES) {
        int d = dst[e];
        unsafeAtomicAdd(&deg[d], ew[e]);
        atomicAdd(&cnt[d], 1);
    }
}

__global__ void make_dinv_kernel(const float* __restrict__ deg,
                                 float* __restrict__ dinv) {
    int i = blockIdx.x * blockDim.x + threadIdx.x;
    if (i < N_NODES) {
        float d = deg[i];
        dinv[i] = d > 0.0f ? rsqrtf(d) : 0.0f;
    }
}

// ---------------- exclusive scan of in-degree counts (CSR offsets) ----------

__global__ __launch_bounds__(256)
void scan1_kernel(const int* __restrict__ cnt,
                  int* __restrict__ offs,
                  int* __restrict__ blockSums) {
    __shared__ int sh[256];
    const int tid = threadIdx.x;
    const int i = blockIdx.x * 256 + tid;
    int v = (i < N_NODES) ? cnt[i] : 0;
    sh[tid] = v;
    __syncthreads();
    for (int d = 1; d < 256; d <<= 1) {
        int t = (tid >= d) ? sh[tid - d] : 0;
        __syncthreads();
        sh[tid] += t;
        __syncthreads();
    }
    if (i < N_NODES) offs[i] = sh[tid] - v;            // exclusive
    if (tid == 255) blockSums[blockIdx.x] = sh[255];   // block total
}

__global__ __launch_bounds__(512)
void scan2_kernel(int* __restrict__ blockSums) {
    __shared__ int sh[512];
    const int tid = threadIdx.x;
    int v = (tid < SCAN_BLOCKS) ? blockSums[tid] : 0;
    sh[tid] = v;
    __syncthreads();
    for (int d = 1; d < 512; d <<= 1) {
        int t = (tid >= d) ? sh[tid - d] : 0;
        __syncthreads();
        sh[tid] += t;
        __syncthreads();
    }
    if (tid < SCAN_BLOCKS) blockSums[tid] = sh[tid] - v;  // exclusive
}

__global__ void scan3_kernel(int* __restrict__ offs,
                             const int* __restrict__ blockSums,
                             int* __restrict__ cursor) {
    int i = blockIdx.x * blockDim.x + threadIdx.x;
    if (i < N_NODES) {
        int o = offs[i] + blockSums[i >> 8];
        offs[i] = o;
        cursor[i] = o;
    }
    if (i == 0) offs[N_NODES] = N_EDGES;
}

// ---------------- CSR fill: bucket edges by dst, fold in edge norm ----------

__global__ void csr_fill_kernel(const int* __restrict__ src,
                                const int* __restrict__ dst,
                                const float* __restrict__ ew,
                                const float* __restrict__ dinv,
                                int* __restrict__ cursor,
                                int* __restrict__ csr_src,
                                float* __restrict__ csr_w) {
    int e = blockIdx.x * blockDim.x + threadIdx.x;
    if (e < N_EDGES) {
        int s = src[e], d = dst[e];
        int pos = atomicAdd(&cursor[d], 1);
        csr_src[pos] = s;
        csr_w[pos] = dinv[s] * ew[e] * dinv[d];
    }
}

// ---------------- dense transform: H = X @ W  (fp32 WMMA) ----------------
// Wave w of each 256-thread block computes the 16x16 tile at rows
// [16*bid, +16), cols [16*w, +16); 32 x v_wmma_f32_16x16x4_f32 per wave.
__global__ __launch_bounds__(256)
void gemm_wmma_f32(const float* __restrict__ X,
                   const float* __restrict__ W,
                   float* __restrict__ H) {
    const int wave = threadIdx.x >> 5;
    const int lane = threadIdx.x & 31;
    const int half = lane >> 4;
    const int r    = lane & 15;
    const int row0 = blockIdx.x * 16;
    const int col0 = wave * 16;

    v8f c = {};
    const float* xrow = X + (size_t)(row0 + r) * DCH;

#pragma unroll
    for (int k = 0; k < DCH; k += 4) {
        const int ka = k + 2 * half;
        v2f a, b;
        a.x = xrow[ka];
        a.y = xrow[ka + 1];
        b.x = W[(size_t)ka * DCH + col0 + r];
        b.y = W[(size_t)(ka + 1) * DCH + col0 + r];
        c = __builtin_amdgcn_wmma_f32_16x16x4_f32(
                false, a, false, b, (short)0, c, false, false);
    }

    float* hout = H + (size_t)(row0 + 8 * half) * DCH + col0 + r;
#pragma unroll
    for (int v = 0; v < 8; ++v) hout[(size_t)v * DCH] = c[v];
}

// ---------------- fused pull-aggregation + self-loop + bias + BN + ReLU -----
// One wave per destination node; lane l owns channels [4l, 4l+4) in a float4
// register accumulator. Gathers of h[src] hit L2 (h = 51MB < 192MB L2).
// No atomics, no agg buffer, epilogue fused.
__global__ __launch_bounds__(256)
void aggregate_bn_relu_kernel(const float* __restrict__ H,
                              const int* __restrict__ offs,
                              const int* __restrict__ csr_src,
                              const float* __restrict__ csr_w,
                              const float* __restrict__ dinv,
                              const float* __restrict__ bias,
                              const float* __restrict__ gamma,
                              const float* __restrict__ beta,
                              const float* __restrict__ mean,
                              const float* __restrict__ var,
                              float* __restrict__ out) {
    const int node = blockIdx.x * 8 + (threadIdx.x >> 5);
    if (node >= N_NODES) return;
    const int lane = threadIdx.x & 31;

    const int beg = offs[node];
    const int end = offs[node + 1];
    const float di = dinv[node];

    // self-loop contribution: norm_ii = dinv^2 * h[node]
    float4 hv = ((const float4*)(H + (size_t)node * DCH))[lane];
    const float sw = di * di;
    float4 acc = make_float4(hv.x * sw, hv.y * sw, hv.z * sw, hv.w * sw);

    for (int j = beg; j < end; ++j) {
        const int   s = csr_src[j];   // wave-uniform -> scalar load
        const float w = csr_w[j];
        float4 v = ((const float4*)(H + (size_t)s * DCH))[lane];
        acc.x += w * v.x;
        acc.y += w * v.y;
        acc.z += w * v.z;
        acc.w += w * v.w;
    }

    const int c0 = lane * 4;
    float4 o;
    {
        float y0 = (acc.x + bias[c0+0] - mean[c0+0]) * rsqrtf(var[c0+0] + BN_EPS) * gamma[c0+0] + beta[c0+0];
        float y1 = (acc.y + bias[c0+1] - mean[c0+1]) * rsqrtf(var[c0+1] + BN_EPS) * gamma[c0+1] + beta[c0+1];
        float y2 = (acc.z + bias[c0+2] - mean[c0+2]) * rsqrtf(var[c0+2] + BN_EPS) * gamma[c0+2] + beta[c0+2];
        float y3 = (acc.w + bias[c0+3] - mean[c0+3]) * rsqrtf(var[c0+3] + BN_EPS) * gamma[c0+3] + beta[c0+3];
        o = make_float4(fmaxf(y0, 0.f), fmaxf(y1, 0.f), fmaxf(y2, 0.f), fmaxf(y3, 0.f));
    }
    ((float4*)(out + (size_t)node * DCH))[lane] = o;
}

// ---------------- driver ----------------

static inline size_t align256(size_t x) { return (x + 255) & ~(size_t)255; }

extern "C" void kernel_launch(void* const* d_in, const int* in_sizes, int n_in,
                              void* d_out, int out_size, void* d_ws, size_t ws_size,
                              hipStream_t stream) {
    const float* x     = (const float*)d_in[0];
    const int*   ei    = (const int*)  d_in[1];   // [2, E]
    const float* ew    = (const float*)d_in[2];
    const float* W     = (const float*)d_in[3];   // [3,128,128]
    const float* bias  = (const float*)d_in[4];   // [3,128]
    const float* gamma = (const float*)d_in[5];
    const float* beta  = (const float*)d_in[6];
    const float* rmean = (const float*)d_in[7];
    const float* rvar  = (const float*)d_in[8];
    float* out = (float*)d_out;

    const int* src = ei;
    const int* dst = ei + N_EDGES;

    // workspace carve (256B-aligned chunks)
    char* p = (char*)d_ws;
    float* deg       = (float*)p; p += align256(sizeof(float) * N_NODES);
    float* dinv      = (float*)p; p += align256(sizeof(float) * N_NODES);
    int*   cnt       = (int*)  p; p += align256(sizeof(int)   * N_NODES);
    int*   offs      = (int*)  p; p += align256(sizeof(int)   * (N_NODES + 1));
    int*   cursor    = (int*)  p; p += align256(sizeof(int)   * N_NODES);
    int*   blockSums = (int*)  p; p += align256(sizeof(int)   * 512);
    int*   csr_src   = (int*)  p; p += align256(sizeof(int)   * N_EDGES);
    float* csr_w     = (float*)p; p += align256(sizeof(float) * N_EDGES);
    float* h         = (float*)p; p += align256(sizeof(float) * (size_t)N_NODES * DCH);
    float* xbuf      = (float*)p;

    const int nodeBlocks = (N_NODES + 255) / 256;
    const int edgeBlocks = (N_EDGES + 255) / 256;

    // --- one-time: normalization + dst-CSR build ---
    init_kernel<<<nodeBlocks, 256, 0, stream>>>(deg, cnt);
    deg_count_kernel<<<edgeBlocks, 256, 0, stream>>>(dst, ew, deg, cnt);
    make_dinv_kernel<<<nodeBlocks, 256, 0, stream>>>(deg, dinv);
    scan1_kernel<<<SCAN_BLOCKS, 256, 0, stream>>>(cnt, offs, blockSums);
    scan2_kernel<<<1, 512, 0, stream>>>(blockSums);
    scan3_kernel<<<nodeBlocks, 256, 0, stream>>>(offs, blockSums, cursor);
    csr_fill_kernel<<<edgeBlocks, 256, 0, stream>>>(src, dst, ew, dinv,
                                                    cursor, csr_src, csr_w);

    // --- layers; ping-pong so the last layer lands in d_out ---
    const float* xin = x;
    float* layer_out[NLAYERS] = { out, xbuf, out };
    for (int n = 0; n < NLAYERS; ++n) {
        gemm_wmma_f32<<<N_NODES / 16, 256, 0, stream>>>(
            xin, W + (size_t)n * DCH * DCH, h);
        aggregate_bn_relu_kernel<<<(N_NODES + 7) / 8, 256, 0, stream>>>(
            h, offs, csr_src, csr_w, dinv,
            bias + n * DCH, gamma + n * DCH, beta + n * DCH,
            rmean + n * DCH, rvar + n * DCH, layer_out[n]);
        xin = layer_out[n];
    }
}